// PointMatching_40011915329643
// MI455X (gfx1250) — compile-verified
//
#include <hip/hip_runtime.h>
#include <hip/hip_bf16.h>

// Problem constants (fixed by the reference harness)
#define B_     1024
#define N_     128
#define M_     128
#define K_     3
#define SLOTS  (B_ * N_ * K_)   /* 393216 output correspondence slots */
#define MAXPB  (N_ * K_)        /* max matches per batch = 384 */

typedef unsigned int v4u __attribute__((ext_vector_type(4)));
typedef int          v8i __attribute__((ext_vector_type(8)));
typedef int          v4i __attribute__((ext_vector_type(4)));

__device__ __forceinline__ void top3_update(float v, float& a, float& b, float& c) {
  if (v > c) {
    if (v > b) {
      c = b;
      if (v > a) { b = a; a = v; } else { b = v; }
    } else {
      c = v;
    }
  }
}

// ---------------------------------------------------------------------------
// Kernel 1: per-batch mutual-top3 matching.
//   - TDM (tensor_load_to_lds) stages the 128x128 f32 score tile into LDS
//   - exp transform, row/col top-3, mask+threshold, per-batch compaction
// ---------------------------------------------------------------------------
__global__ __launch_bounds__(128) void pm_batch_kernel(
    const float* __restrict__ score,            // [B,N,M]
    const unsigned char* __restrict__ refMask,  // [B,N]
    const unsigned char* __restrict__ srcMask,  // [B,M]
    int*   __restrict__ wsLin,                  // [B,MAXPB] local linear idx r*128+c
    float* __restrict__ wsScore,                // [B,MAXPB] exp(score) at match
    int*   __restrict__ wsCnt,                  // [B]
    int*   __restrict__ wsFlag)                 // [1] corr[0,0,0] pad flag
{
  __shared__ float tile[N_ * M_];               // 64 KB (of 320 KB/WGP)
  __shared__ float colThird[M_];
  __shared__ unsigned char sMaskSh[M_];
  __shared__ int scanBuf[N_];

  const int b = blockIdx.x;
  const int t = threadIdx.x;

  // ---- Tensor Data Mover: async DMA of the 64KB tile into LDS (wave 0) ----
  if (t < warpSize) {
    unsigned long long gaddr =
        (unsigned long long)(uintptr_t)(score + (size_t)b * (N_ * M_));
    unsigned int alo = (unsigned int)__builtin_amdgcn_readfirstlane(
        (int)(gaddr & 0xFFFFFFFFull));
    unsigned int ahi = (unsigned int)__builtin_amdgcn_readfirstlane(
        (int)((gaddr >> 32) & 0x1FFFFFFull));
    unsigned int ldsOff = (unsigned int)__builtin_amdgcn_readfirstlane(
        (int)(unsigned int)(uintptr_t)&tile[0]);

    v4u g0;
    g0.x = 1u;                     // count=1 (valid user descriptor)
    g0.y = ldsOff;                 // lds_addr (bytes)
    g0.z = alo;                    // global_addr[31:0]
    g0.w = ahi | (2u << 30);       // global_addr[56:32] | type=2 ("image")

    v8i g1;
    g1[0] = (int)(2u << 16);       // workgroup_mask=0, data_size=2 (4 bytes)
    g1[1] = (int)(128u << 16);     // tensor_dim0 = 128 (bits 79:48, lo16)
    g1[2] = (int)(128u << 16);     // tensor_dim0 hi=0 | tensor_dim1 = 128 lo16
    g1[3] = (int)(128u << 16);     // tensor_dim1 hi=0 | tile_dim0 = 128
    g1[4] = 128;                   // tile_dim1 = 128, tile_dim2 = 0
    g1[5] = 128;                   // tensor_dim0_stride = 128 (lo32)
    g1[6] = (int)(16384u << 16);   // dim0_stride hi=0 | tensor_dim1_stride lo16
    g1[7] = 0;                     // tensor_dim1_stride hi
    v4i gz = {0, 0, 0, 0};
#if __clang_major__ >= 23
    v8i gz8 = {0, 0, 0, 0, 0, 0, 0, 0};
    __builtin_amdgcn_tensor_load_to_lds(g0, g1, gz, gz, gz8, 0);
#else
    __builtin_amdgcn_tensor_load_to_lds(g0, g1, gz, gz, 0);
#endif
    __builtin_amdgcn_s_wait_tensorcnt(0);
  }
  __syncthreads();

  // ---- s = exp(score): column-strided (bank = t%64, conflict-free) ----
  for (int i = 0; i < N_; ++i) {
    int idx = i * M_ + t;
    tile[idx] = expf(tile[idx]);
  }
  __syncthreads();

  // ---- row top-3 for row t (rotated reads: bank=(t+i)%64, conflict-free) --
  float ra = -1.0f, rb = -1.0f, rcv = -1.0f;   // exp(x) > 0, so -1 is -inf
  {
    const int base = t * M_;
    for (int i = 0; i < M_; ++i) {
      int c = (i + t) & (M_ - 1);
      top3_update(tile[base + c], ra, rb, rcv);
    }
  }
  const float rowThird = rcv;

  // ---- column top-3 for column t (conflict-free) ----
  float ca = -1.0f, cb = -1.0f, ccv = -1.0f;
  for (int i = 0; i < N_; ++i) top3_update(tile[i * M_ + t], ca, cb, ccv);
  colThird[t] = ccv;
  sMaskSh[t]  = srcMask[(size_t)b * M_ + t];
  __syncthreads();

  // ---- mutual top-3 AND masks AND (s > 0.05) -> 128-bit row hit mask ----
  const bool rm = refMask[(size_t)b * N_ + t] != 0;
  unsigned long long mlo = 0ull, mhi = 0ull;
  if (rm) {
    const int base = t * M_;
    for (int i = 0; i < M_; ++i) {
      int c = (i + t) & (M_ - 1);
      float v = tile[base + c];
      bool cor = (v > 0.05f) && (v >= rowThird) && (v >= colThird[c]) &&
                 (sMaskSh[c] != 0);
      unsigned long long bit = cor ? 1ull : 0ull;
      if (c < 64) mlo |= (bit << c);
      else        mhi |= (bit << (c - 64));
    }
  }
  int cnt = __popcll(mlo) + __popcll(mhi);

  // ---- exclusive scan over the 128 rows (Hillis-Steele in LDS) ----
  scanBuf[t] = cnt;
  __syncthreads();
  for (int d = 1; d < N_; d <<= 1) {
    int v = (t >= d) ? scanBuf[t - d] : 0;
    __syncthreads();
    scanBuf[t] += v;
    __syncthreads();
  }
  int off = scanBuf[t] - cnt;                 // batch-local exclusive offset
  if (t == N_ - 1) wsCnt[b] = scanBuf[N_ - 1];
  if (b == 0 && t == 0) wsFlag[0] = (int)(mlo & 1ull);  // corr[0,0,0]

  // ---- emit matches in ascending-column order (reference nonzero order) ---
  const int base = b * MAXPB;
  unsigned long long m = mlo;
  int cbase = 0;
  for (int half = 0; half < 2; ++half) {
    while (m) {
      int c = __builtin_ctzll(m);
      m &= (m - 1ull);
      int col = cbase + c;
      wsLin[base + off]   = t * M_ + col;
      wsScore[base + off] = tile[t * M_ + col];
      ++off;
    }
    m = mhi;
    cbase = 64;
  }
}

// ---------------------------------------------------------------------------
// Kernel 2: exclusive scan of the 1024 batch counts (global compaction order)
// ---------------------------------------------------------------------------
__global__ __launch_bounds__(1024) void pm_scan_kernel(
    const int* __restrict__ wsCnt, int* __restrict__ wsOff) {
  __shared__ int sh[B_];
  int t = threadIdx.x;
  int v0 = wsCnt[t];
  sh[t] = v0;
  __syncthreads();
  for (int d = 1; d < B_; d <<= 1) {
    int v = (t >= d) ? sh[t - d] : 0;
    __syncthreads();
    sh[t] += v;
    __syncthreads();
  }
  wsOff[t] = sh[t] - v0;
}

// ---------------------------------------------------------------------------
// Kernel 3: fill all S slots with the pad value. Reference pads lin=0, so pad
// slots replicate entry (b=0,r=0,c=0) iff corr[0,0,0] else zeros.
// ---------------------------------------------------------------------------
__global__ void pm_pad_kernel(const float* __restrict__ refPts,
                              const float* __restrict__ srcPts,
                              const int* __restrict__ refIdx,
                              const int* __restrict__ srcIdx,
                              const float* __restrict__ score,
                              const int* __restrict__ wsFlag,
                              float* __restrict__ out) {
  int j = blockIdx.x * blockDim.x + threadIdx.x;
  if (j >= SLOTS) return;
  const bool v = wsFlag[0] != 0;
  const float f = v ? 1.0f : 0.0f;
  float p0 = refPts[0] * f, p1 = refPts[1] * f, p2 = refPts[2] * f;
  float q0 = srcPts[0] * f, q1 = srcPts[1] * f, q2 = srcPts[2] * f;
  float ri = v ? (float)refIdx[0] : 0.0f;
  float si = v ? (float)srcIdx[0] : 0.0f;
  float sc = v ? expf(score[0]) : 0.0f;
  out[(size_t)j * 3 + 0] = p0;
  out[(size_t)j * 3 + 1] = p1;
  out[(size_t)j * 3 + 2] = p2;
  float* o2 = out + (size_t)3 * SLOTS;
  o2[(size_t)j * 3 + 0] = q0;
  o2[(size_t)j * 3 + 1] = q1;
  o2[(size_t)j * 3 + 2] = q2;
  out[(size_t)6 * SLOTS + j] = ri;
  out[(size_t)7 * SLOTS + j] = si;
  out[(size_t)8 * SLOTS + j] = sc;
}

// ---------------------------------------------------------------------------
// Kernel 4: scatter valid matches into the first `total` slots.
// ---------------------------------------------------------------------------
__global__ __launch_bounds__(128) void pm_scatter_kernel(
    const float* __restrict__ refPts, const float* __restrict__ srcPts,
    const int* __restrict__ refIdx, const int* __restrict__ srcIdx,
    const int* __restrict__ wsLin, const float* __restrict__ wsScore,
    const int* __restrict__ wsCnt, const int* __restrict__ wsOff,
    float* __restrict__ out) {
  int b = blockIdx.x;
  int t = threadIdx.x;
  int cnt  = wsCnt[b];
  int base = wsOff[b];
  for (int e = t; e < cnt; e += blockDim.x) {
    int lin = wsLin[b * MAXPB + e];
    int r = lin >> 7;
    int c = lin & (M_ - 1);
    int j = base + e;
    const float* rp = refPts + ((size_t)b * N_ + r) * 3;
    const float* sp = srcPts + ((size_t)b * M_ + c) * 3;
    out[(size_t)j * 3 + 0] = rp[0];
    out[(size_t)j * 3 + 1] = rp[1];
    out[(size_t)j * 3 + 2] = rp[2];
    float* o2 = out + (size_t)3 * SLOTS;
    o2[(size_t)j * 3 + 0] = sp[0];
    o2[(size_t)j * 3 + 1] = sp[1];
    o2[(size_t)j * 3 + 2] = sp[2];
    out[(size_t)6 * SLOTS + j] = (float)refIdx[(size_t)b * N_ + r];
    out[(size_t)7 * SLOTS + j] = (float)srcIdx[(size_t)b * M_ + c];
    out[(size_t)8 * SLOTS + j] = wsScore[b * MAXPB + e];
  }
}

extern "C" void kernel_launch(void* const* d_in, const int* in_sizes, int n_in,
                              void* d_out, int out_size, void* d_ws, size_t ws_size,
                              hipStream_t stream) {
  (void)in_sizes; (void)n_in; (void)out_size; (void)ws_size;
  const float*         refPts  = (const float*)d_in[0];          // [B,N,3]
  const float*         srcPts  = (const float*)d_in[1];          // [B,M,3]
  const unsigned char* refMask = (const unsigned char*)d_in[2];  // [B,N] bool
  const unsigned char* srcMask = (const unsigned char*)d_in[3];  // [B,M] bool
  const int*           refIdx  = (const int*)d_in[4];            // [B,N]
  const int*           srcIdx  = (const int*)d_in[5];            // [B,M]
  const float*         score   = (const float*)d_in[6];          // [B,N,M]
  // d_in[7] global_scores: unused (use_global_score=False)
  float* out = (float*)d_out;

  // Workspace layout (all offsets deterministic; initialized every call)
  char*  ws      = (char*)d_ws;
  int*   wsLin   = (int*)ws;                                    // B*MAXPB ints
  float* wsScore = (float*)(ws + (size_t)B_ * MAXPB * 4);       // B*MAXPB floats
  int*   wsCnt   = (int*)(ws + (size_t)B_ * MAXPB * 8);         // B ints
  int*   wsOff   = wsCnt + B_;                                  // B ints
  int*   wsFlag  = wsOff + B_;                                  // 1 int

  pm_batch_kernel<<<B_, N_, 0, stream>>>(score, refMask, srcMask,
                                         wsLin, wsScore, wsCnt, wsFlag);
  pm_scan_kernel<<<1, B_, 0, stream>>>(wsCnt, wsOff);
  pm_pad_kernel<<<(SLOTS + 255) / 256, 256, 0, stream>>>(
      refPts, srcPts, refIdx, srcIdx, score, wsFlag, out);
  pm_scatter_kernel<<<B_, N_, 0, stream>>>(refPts, srcPts, refIdx, srcIdx,
                                           wsLin, wsScore, wsCnt, wsOff, out);
}